// encoder_9680856285475
// MI455X (gfx1250) — compile-verified
//
#include <hip/hip_runtime.h>

#define NN  100000
#define EE  1600000
#define FIN 128
#define HH  128
#define KIN 512   // (K+1)*128 concat width

typedef float v2f __attribute__((ext_vector_type(2)));
typedef float v8f __attribute__((ext_vector_type(8)));

// ---------------- degree / norm ----------------
__global__ void deg_kernel(const int* __restrict__ dst, float* __restrict__ deg) {
    int e = blockIdx.x * blockDim.x + threadIdx.x;
    if (e < EE) atomicAdd(&deg[dst[e]], 1.0f);
}

__global__ void norm_kernel(float* __restrict__ norm) {
    int i = blockIdx.x * blockDim.x + threadIdx.x;
    if (i < NN) {
        float d = norm[i];
        norm[i] = rsqrtf(fmaxf(d, 1.0f));   // clip(deg,1)^-0.5
    }
}

// ---------------- copy h -> feats slot 0 (stride 512) ----------------
__global__ void copy_slot0(const float* __restrict__ h, float* __restrict__ F) {
    int t = blockIdx.x * blockDim.x + threadIdx.x;
    if (t < NN * FIN) {
        int node = t >> 7;
        int f    = t & 127;
        F[(size_t)node * KIN + f] = h[t];
    }
}

// ---------------- one hop: accum[dst] += hin[src] * norm[src] ----------------
// hin/accum are slot pointers into the [N,512] feats buffer (row stride 512).
// 32 lanes per edge, float4 per lane: coalesced 512B gather per edge.
__global__ void spmm_hop(const float* __restrict__ hin, float* __restrict__ accum,
                         const float* __restrict__ norm,
                         const int* __restrict__ src, const int* __restrict__ dst) {
    int t = blockIdx.x * blockDim.x + threadIdx.x;
    int e = t >> 5;
    int g = t & 31;
    if (e < EE) {
        int s = src[e], d = dst[e];
        float ns = norm[s];
        const float4 v = *reinterpret_cast<const float4*>(hin + (size_t)s * KIN + g * 4);
        float* pd = accum + (size_t)d * KIN + g * 4;
        atomicAdd(pd + 0, v.x * ns);
        atomicAdd(pd + 1, v.y * ns);
        atomicAdd(pd + 2, v.z * ns);
        atomicAdd(pd + 3, v.w * ns);
    }
}

// ---------------- post-aggregation scale by norm[node] (in place) ----------------
__global__ void scale_slot(float* __restrict__ slot, const float* __restrict__ norm) {
    int t = blockIdx.x * blockDim.x + threadIdx.x;
    if (t < NN * FIN) {
        int node = t >> 7;
        int f    = t & 127;
        slot[(size_t)node * KIN + f] *= norm[node];
    }
}

// ---------------- WMMA GEMM: out[N,128] = relu(F[N,512] @ W[512,128] + b) ----------------
// Block = 256 threads = 8 waves; block owns 16 rows; wave w owns cols [16w,16w+16).
// fp32 WMMA 16x16x4. A/B per-lane element j = K index (k0 + j + 2*half) per ISA tables.
__global__ __launch_bounds__(256) void tag_gemm(const float* __restrict__ F,
                                                const float* __restrict__ W,
                                                const float* __restrict__ bias,
                                                float* __restrict__ out) {
    __shared__ float As[16][KIN + 4];            // pad 516: column reads hit distinct banks
    const int row0 = blockIdx.x * 16;

    // stage 16x512 A-tile, coalesced float4 loads
    for (int i = threadIdx.x; i < 16 * (KIN / 4); i += 256) {
        int r  = i / (KIN / 4);
        int c4 = (i % (KIN / 4)) * 4;
        float4 v = *reinterpret_cast<const float4*>(F + (size_t)(row0 + r) * KIN + c4);
        *reinterpret_cast<float4*>(&As[r][c4]) = v;
    }
    __syncthreads();

    const int wave = threadIdx.x >> 5;
    const int lane = threadIdx.x & 31;
    const int half = lane >> 4;      // 0: K-even pair / rows M,N 0-7 ; 1: +2 K / +8 M
    const int mn   = lane & 15;
    const int col0 = wave * 16;

    v8f acc = {0.f, 0.f, 0.f, 0.f, 0.f, 0.f, 0.f, 0.f};
    for (int k0 = 0; k0 < KIN; k0 += 4) {
        const int kk = k0 + half * 2;
        v2f a, b;
        a.x = As[mn][kk];
        a.y = As[mn][kk + 1];
        b.x = W[(size_t)kk * HH + col0 + mn];
        b.y = W[(size_t)(kk + 1) * HH + col0 + mn];
        acc = __builtin_amdgcn_wmma_f32_16x16x4_f32(false, a, false, b,
                                                    (short)0, acc, false, false);
    }

    const float bv = bias[col0 + mn];
#pragma unroll
    for (int r = 0; r < 8; ++r) {
        float v = acc[r] + bv;
        out[(size_t)(row0 + r + half * 8) * HH + col0 + mn] = v > 0.f ? v : 0.f;
    }
}

// ---------------- launch ----------------
extern "C" void kernel_launch(void* const* d_in, const int* in_sizes, int n_in,
                              void* d_out, int out_size, void* d_ws, size_t ws_size,
                              hipStream_t stream) {
    const float* x  = (const float*)d_in[0];
    const int*   ei = (const int*)d_in[1];     // [2,E] row-major: src then dst
    const float* W1 = (const float*)d_in[2];
    const float* b1 = (const float*)d_in[3];
    const float* W2 = (const float*)d_in[4];
    const float* b2 = (const float*)d_in[5];
    const int* src = ei;
    const int* dst = ei + EE;

    float* norm = (float*)d_ws;                      // [N]
    float* F    = norm + NN;                         // [N,512] feats (reused both layers)
    float* h1   = F + (size_t)NN * KIN;              // [N,128] layer-1 output
    float* outp = (float*)d_out;                     // [N,128]

    const dim3 b256(256);
    const int gE    = (EE + 255) / 256;
    const int gN    = (NN + 255) / 256;
    const int gNF   = (NN * FIN + 255) / 256;
    const int gHop  = (EE * 32 + 255) / 256;         // 200000 blocks
    const int gGemm = NN / 16;                       // 6250 (exact)

    hipMemsetAsync(norm, 0, (size_t)NN * sizeof(float), stream);
    deg_kernel<<<gE, b256, 0, stream>>>(dst, norm);
    norm_kernel<<<gN, b256, 0, stream>>>(norm);

    for (int layer = 0; layer < 2; ++layer) {
        hipMemsetAsync(F, 0, (size_t)NN * KIN * sizeof(float), stream);
        copy_slot0<<<gNF, b256, 0, stream>>>(layer == 0 ? x : h1, F);
        for (int k = 1; k <= 3; ++k) {
            spmm_hop<<<gHop, b256, 0, stream>>>(F + (k - 1) * FIN, F + k * FIN,
                                                norm, src, dst);
            scale_slot<<<gNF, b256, 0, stream>>>(F + k * FIN, norm);
        }
        if (layer == 0)
            tag_gemm<<<gGemm, b256, 0, stream>>>(F, W1, b1, h1);
        else
            tag_gemm<<<gGemm, b256, 0, stream>>>(F, W2, b2, outp);
    }
}